// WienerDeconvolution_58686433132996
// MI455X (gfx1250) — compile-verified
//
#include <hip/hip_runtime.h>
#include <math.h>

// ---------------------------------------------------------------------------
// Wiener deconvolution (edgetaper + FFT-domain solve) for MI455X / gfx1250.
// All 2-D FFTs (N = 542 = 2*271, 271 prime -> radix FFT is awkward) are done
// as dense DFT matmuls  fft2(X) = W * X * W  on the f32 WMMA matrix cores.
// ---------------------------------------------------------------------------

typedef __attribute__((ext_vector_type(2))) float v2f;
typedef __attribute__((ext_vector_type(8))) float v8f;

#define H0    512
#define NPAD  542
#define NN    (NPAD * NPAD)
#define BB    8
#define CC    3
#define NIMG  (BB * CC)
#define KH    31
#define PH    15
#define NF    8
#define PI2   6.28318530717958647692f

__device__ __forceinline__ v8f wmma4(v2f a, v2f b, v8f c) {
  // D(16x16,f32) += A(16x4,f32) * B(4x16,f32); wave32, 8-arg VOP3P form.
  return __builtin_amdgcn_wmma_f32_16x16x4_f32(false, a, false, b,
                                               (short)0, c, false, false);
}

// ---------------------------------------------------------------------------
// Batched complex GEMM: D = scale * op(A) * op(B), all square N x N row-major.
// Compile-time variants (no branches / accumulator phi-copies in the K loop):
//   HAS_AI/HAS_BI : operand has an imaginary part
//   HAS_DI        : emit imaginary output (false => ifft2(.).real epilogue)
// conjA/conjB fold the conjugation sign into LDS staging (f32 WMMA NEG bits
// are C-negate only, so A/B negation must happen in registers/LDS).
// Tiling: 64x64 output tile per 256-thread block; 8 waves; each wave owns two
// vertically stacked 16x16 WMMA subtiles; K staged 16 wide through LDS.
// N is even and all buffers 8B-aligned => float2 global staging is safe.
// ---------------------------------------------------------------------------
template <bool HAS_AI, bool HAS_BI, bool HAS_DI>
__global__ __launch_bounds__(256)
void cgemm_kernel(const float* __restrict__ Ar, const float* __restrict__ Ai, long sA_,
                  const float* __restrict__ Br, const float* __restrict__ Bi, long sB_,
                  float* __restrict__ Dr, float* __restrict__ Di, long sD_,
                  int n, float scale, int conjA, int conjB) {
  __shared__ float sAr[64][18];   // stride 18: even (8B-aligned pairs), 16 rows
  __shared__ float sAi[64][18];   //   at stride 18 mod 64 banks -> conflict-free
  __shared__ float sBr[16][66];
  __shared__ float sBi[16][66];

  const int  tileM = blockIdx.x * 64;
  const int  tileN = blockIdx.y * 64;
  const long img   = blockIdx.z;
  const float sgA  = conjA ? -1.f : 1.f;
  const float sgB  = conjB ? -1.f : 1.f;

  const float* arp = Ar + img * sA_;
  const float* aip = HAS_AI ? (Ai + img * sA_) : nullptr;
  const float* brp = Br + img * sB_;
  const float* bip = HAS_BI ? (Bi + img * sB_) : nullptr;

  const int wave = threadIdx.x >> 5;
  const int lane = threadIdx.x & 31;
  const int h    = lane >> 4;          // half-wave select (K / M-row split)
  const int l16  = lane & 15;
  const int colT = (wave & 3) * 16;    // subtile column inside 64-wide tile
  const int rowG = (wave >> 2) * 32;   // subtile row group inside 64-tall tile

  v8f accR[2], accI[2];
  const v8f z8 = {0.f, 0.f, 0.f, 0.f, 0.f, 0.f, 0.f, 0.f};
  accR[0] = z8; accR[1] = z8; accI[0] = z8; accI[1] = z8;

  for (int k0 = 0; k0 < n; k0 += 16) {
    // stage A tile: 64 rows x 16 K-cols as float2 (512 pairs / 256 threads)
    for (int idx = threadIdx.x; idx < 64 * 8; idx += 256) {
      const int r = idx >> 3, c2 = (idx & 7) * 2;
      const int gr = tileM + r, gc = k0 + c2;
      v2f vr = {0.f, 0.f}, vi = {0.f, 0.f};
      if (gr < n && gc < n) {  // n even, gc even => pair fully in range
        const long o = (long)gr * n + gc;
        vr = *(const v2f*)(arp + o);
        if constexpr (HAS_AI) { vi = *(const v2f*)(aip + o); vi = sgA * vi; }
      }
      *(v2f*)&sAr[r][c2] = vr;
      if constexpr (HAS_AI) *(v2f*)&sAi[r][c2] = vi;
    }
    // stage B tile: 16 K-rows x 64 cols as float2
    for (int idx = threadIdx.x; idx < 16 * 32; idx += 256) {
      const int r = idx >> 5, c2 = (idx & 31) * 2;
      const int gr = k0 + r, gc = tileN + c2;
      v2f vr = {0.f, 0.f}, vi = {0.f, 0.f};
      if (gr < n && gc < n) {
        const long o = (long)gr * n + gc;
        vr = *(const v2f*)(brp + o);
        if constexpr (HAS_BI) { vi = *(const v2f*)(bip + o); vi = sgB * vi; }
      }
      *(v2f*)&sBr[r][c2] = vr;
      if constexpr (HAS_BI) *(v2f*)&sBi[r][c2] = vi;
    }
    __syncthreads();

#pragma unroll
    for (int step = 0; step < 4; ++step) {
      const int kk = step * 4;
      // B fragment (4x16 f32): VGPR0 = K(kk+2h), VGPR1 = K(kk+2h+1), lane%16 = N
      v2f brf; brf.x = sBr[kk + 2 * h][colT + l16]; brf.y = sBr[kk + 2 * h + 1][colT + l16];
      v2f bif = {0.f, 0.f}, bnf = {0.f, 0.f};
      if constexpr (HAS_BI) {
        bif.x = sBi[kk + 2 * h][colT + l16]; bif.y = sBi[kk + 2 * h + 1][colT + l16];
        bnf = -bif;  // for  Dr -= Ai_eff * Bi_eff
      }
      // A fragments (16x4 f32): lane%16 = M, VGPRs hold K(kk+2h), K(kk+2h+1)
      const int row0 = rowG + l16, row1 = rowG + 16 + l16;
      v2f arf0; arf0.x = sAr[row0][kk + 2 * h]; arf0.y = sAr[row0][kk + 2 * h + 1];
      v2f arf1; arf1.x = sAr[row1][kk + 2 * h]; arf1.y = sAr[row1][kk + 2 * h + 1];
      v2f aif0 = {0.f, 0.f}, aif1 = {0.f, 0.f};
      if constexpr (HAS_AI) {
        aif0.x = sAi[row0][kk + 2 * h]; aif0.y = sAi[row0][kk + 2 * h + 1];
        aif1.x = sAi[row1][kk + 2 * h]; aif1.y = sAi[row1][kk + 2 * h + 1];
      }
      // interleaved so same-accumulator WMMAs are >=4 apart (hazard hiding)
      accR[0] = wmma4(arf0, brf, accR[0]);
      if constexpr (HAS_DI && HAS_BI) accI[0] = wmma4(arf0, bif, accI[0]);
      accR[1] = wmma4(arf1, brf, accR[1]);
      if constexpr (HAS_DI && HAS_BI) accI[1] = wmma4(arf1, bif, accI[1]);
      if constexpr (HAS_AI && HAS_BI) accR[0] = wmma4(aif0, bnf, accR[0]);
      if constexpr (HAS_DI && HAS_AI) accI[0] = wmma4(aif0, brf, accI[0]);
      if constexpr (HAS_AI && HAS_BI) accR[1] = wmma4(aif1, bnf, accR[1]);
      if constexpr (HAS_DI && HAS_AI) accI[1] = wmma4(aif1, brf, accI[1]);
    }
    __syncthreads();
  }

  // store: D 16x16 f32 layout -> row = base + reg + 8*h, col = lane%16
  float* drp = Dr + img * sD_;
  float* dip = HAS_DI ? (Di + img * sD_) : nullptr;
#pragma unroll
  for (int r = 0; r < 2; ++r) {
    const int col = tileN + colT + l16;
    if (col < n) {
      const int baseRow = tileM + rowG + r * 16 + 8 * h;
#pragma unroll
      for (int reg = 0; reg < 8; ++reg) {
        const int row = baseRow + reg;
        if (row < n) {
          const long o = (long)row * n + col;
          drp[o] = scale * accR[r][reg];
          if constexpr (HAS_DI) dip[o] = scale * accI[r][reg];
        }
      }
    }
  }
}

// --------------------------- small helper kernels ---------------------------

// DFT matrix W[u][v] = exp(-2*pi*i*u*v/N); exact phase via integer mod.
__global__ void build_dft_kernel(float* __restrict__ Wr, float* __restrict__ Wi) {
  const int idx = blockIdx.x * blockDim.x + threadIdx.x;
  if (idx >= NN) return;
  const int u = idx / NPAD, v = idx % NPAD;
  const int m = (int)(((long)u * v) % NPAD);
  float s, c;
  __sincosf(-PI2 * (float)m / (float)NPAD, &s, &c);
  Wr[idx] = c;
  Wi[idx] = s;
}

// edge-replicate pad 512 -> 542 (pad 15 on each side)
__global__ void pad_edge_kernel(const float* __restrict__ y, float* __restrict__ X) {
  const int idx = blockIdx.x * blockDim.x + threadIdx.x;
  if (idx >= NIMG * NN) return;
  const int img = idx / NN, uv = idx % NN;
  const int u = uv / NPAD, v = uv % NPAD;
  const int su = min(max(u - PH, 0), H0 - 1);
  const int sv = min(max(v - PH, 0), H0 - 1);
  X[idx] = y[(long)img * H0 * H0 + (long)su * H0 + sv];
}

// psf2otf spatial image for the 31x31 PSF: pad to N, roll by (-15,-15)
__global__ void build_kimg_kernel(const float* __restrict__ k, float* __restrict__ Kimg) {
  const int idx = blockIdx.x * blockDim.x + threadIdx.x;
  if (idx >= BB * NN) return;
  const int b = idx / NN, uv = idx % NN;
  const int u = uv / NPAD, v = uv % NPAD;
  const int i = (u + PH) % NPAD;
  const int j = (v + PH) % NPAD;
  Kimg[idx] = (i < KH && j < KH) ? k[(long)b * KH * KH + i * KH + j] : 0.f;
}

// edgetaper window vectors: 1 - circular-autocorr(proj, len 541)/max, len 542.
// PSF support is 31, so the 541-pt FFT autocorr collapses to 31 direct lags.
__global__ void taper_vecs_kernel(const float* __restrict__ k, float* __restrict__ vtab) {
  __shared__ float proj[KH];
  __shared__ float ac[KH];
  const int b = blockIdx.x, dim = blockIdx.y, t = threadIdx.x;
  const float* kb = k + (long)b * KH * KH;
  if (t < KH) {
    float s = 0.f;
    for (int j = 0; j < KH; ++j) s += (dim == 0) ? kb[t * KH + j] : kb[j * KH + t];
    proj[t] = s;
  }
  __syncthreads();
  if (t < KH) {
    float s = 0.f;
    for (int n2 = 0; n2 + t < KH; ++n2) s += proj[n2] * proj[n2 + t];
    ac[t] = s;
  }
  __syncthreads();
  const float inv0 = 1.f / ac[0];
  for (int i = t; i < NPAD; i += 32) {
    const int ii = (i == NPAD - 1) ? 0 : i;  // z = concat(c, c[0])
    float c;
    if (ii < KH)                          c = ac[ii];
    else if (ii >= (NPAD - 1) - (KH - 1)) c = ac[(NPAD - 1) - ii];
    else                                  c = 0.f;
    vtab[((long)b * 2 + dim) * NPAD + i] = 1.f - c * inv0;
  }
}

// Dg_sum[c][u][v] = sum_f | sum_{i,j} filt[f,c,i,j] e^{-2pi i (u(i-1)+v(j-1))/N} |^2
__global__ __launch_bounds__(256)
void dg_sum_kernel(const float* __restrict__ filt, float* __restrict__ Dg) {
  __shared__ float sf[NF * CC * 9];
  if (threadIdx.x < NF * CC * 9) sf[threadIdx.x] = filt[threadIdx.x];
  __syncthreads();
  const int uv = blockIdx.x * blockDim.x + threadIdx.x;
  if (uv >= NN) return;
  const int u = uv / NPAD, v = uv % NPAD;
  float su, cu, sv, cv;
  __sincosf(PI2 * (float)u / (float)NPAD, &su, &cu);
  __sincosf(PI2 * (float)v / (float)NPAD, &sv, &cv);
  const float rur[3] = {cu, 1.f, cu};
  const float rui[3] = {su, 0.f, -su};
  const float rvr[3] = {cv, 1.f, cv};
  const float rvi[3] = {sv, 0.f, -sv};
  for (int c = 0; c < CC; ++c) {
    float dg = 0.f;
    for (int f = 0; f < NF; ++f) {
      const float* w = &sf[(f * CC + c) * 9];
      float ar = 0.f, ai = 0.f;
#pragma unroll
      for (int i = 0; i < 3; ++i)
#pragma unroll
        for (int j = 0; j < 3; ++j) {
          const float pr = rur[i] * rvr[j] - rui[i] * rvi[j];
          const float pi = rur[i] * rvi[j] + rui[i] * rvr[j];
          const float wv = w[i * 3 + j];
          ar += wv * pr;
          ai += wv * pi;
        }
      dg += ar * ar + ai * ai;
    }
    Dg[(long)c * NN + uv] = dg;
  }
}

// F <- F * Dk  (complex, Dk broadcast over channels)
__global__ void mul_otf_kernel(float* __restrict__ Fr, float* __restrict__ Fi,
                               const float* __restrict__ Dkr, const float* __restrict__ Dki) {
  const int idx = blockIdx.x * blockDim.x + threadIdx.x;
  if (idx >= NIMG * NN) return;
  const int img = idx / NN, uv = idx % NN, b = img / CC;
  const float fr = Fr[idx], fi = Fi[idx];
  const float dr = Dkr[(long)b * NN + uv], di = Dki[(long)b * NN + uv];
  Fr[idx] = fr * dr - fi * di;
  Fi[idx] = fr * di + fi * dr;
}

// x <- alpha*x + (1-alpha)*blur,  alpha = vy[u]*vx[v]
__global__ void blend_kernel(float* __restrict__ X, const float* __restrict__ blur,
                             const float* __restrict__ vtab) {
  const int idx = blockIdx.x * blockDim.x + threadIdx.x;
  if (idx >= NIMG * NN) return;
  const int img = idx / NN, uv = idx % NN, b = img / CC;
  const int u = uv / NPAD, v = uv % NPAD;
  const float a = vtab[((long)b * 2 + 0) * NPAD + u] * vtab[((long)b * 2 + 1) * NPAD + v];
  X[idx] = a * X[idx] + (1.f - a) * blur[idx];
}

// F <- Omega * conj(Dk) * F,  Omega = 1/(|Dk|^2 + exp(lam)*Dg_sum)
__global__ void wiener_kernel(float* __restrict__ Fr, float* __restrict__ Fi,
                              const float* __restrict__ Dkr, const float* __restrict__ Dki,
                              const float* __restrict__ Dg, const float* __restrict__ lam) {
  const int idx = blockIdx.x * blockDim.x + threadIdx.x;
  if (idx >= NIMG * NN) return;
  const int img = idx / NN, uv = idx % NN;
  const int b = img / CC, c = img % CC;
  const float el = __expf(lam[0]);
  const float dr = Dkr[(long)b * NN + uv], di = Dki[(long)b * NN + uv];
  const float w = 1.f / (dr * dr + di * di + el * Dg[(long)c * NN + uv]);
  const float fr = Fr[idx], fi = Fi[idx];
  Fr[idx] = w * (dr * fr + di * fi);
  Fi[idx] = w * (dr * fi - di * fr);
}

// ------------------------------- host driver --------------------------------

static inline void cgemm(const float* Ar, const float* Ai, long sA,
                         const float* Br, const float* Bi, long sB,
                         float* Dr, float* Di, long sD,
                         int nimg, float scale, int conjA, int conjB, hipStream_t st) {
  dim3 g((NPAD + 63) / 64, (NPAD + 63) / 64, nimg);
  const bool ai = (Ai != nullptr), bi = (Bi != nullptr), di = (Di != nullptr);
  if (ai && bi && di)
    cgemm_kernel<true, true, true><<<g, 256, 0, st>>>(Ar, Ai, sA, Br, Bi, sB, Dr, Di, sD, NPAD, scale, conjA, conjB);
  else if (ai && !bi && di)
    cgemm_kernel<true, false, true><<<g, 256, 0, st>>>(Ar, Ai, sA, Br, Bi, sB, Dr, Di, sD, NPAD, scale, conjA, conjB);
  else if (ai && bi && !di)
    cgemm_kernel<true, true, false><<<g, 256, 0, st>>>(Ar, Ai, sA, Br, Bi, sB, Dr, Di, sD, NPAD, scale, conjA, conjB);
  else if (!ai && bi && di)
    cgemm_kernel<false, true, true><<<g, 256, 0, st>>>(Ar, Ai, sA, Br, Bi, sB, Dr, Di, sD, NPAD, scale, conjA, conjB);
  else if (!ai && !bi && di)
    cgemm_kernel<false, false, true><<<g, 256, 0, st>>>(Ar, Ai, sA, Br, Bi, sB, Dr, Di, sD, NPAD, scale, conjA, conjB);
  else if (ai && !bi && !di)
    cgemm_kernel<true, false, false><<<g, 256, 0, st>>>(Ar, Ai, sA, Br, Bi, sB, Dr, Di, sD, NPAD, scale, conjA, conjB);
  else if (!ai && bi && !di)
    cgemm_kernel<false, true, false><<<g, 256, 0, st>>>(Ar, Ai, sA, Br, Bi, sB, Dr, Di, sD, NPAD, scale, conjA, conjB);
  else
    cgemm_kernel<false, false, false><<<g, 256, 0, st>>>(Ar, Ai, sA, Br, Bi, sB, Dr, Di, sD, NPAD, scale, conjA, conjB);
}

extern "C" void kernel_launch(void* const* d_in, const int* in_sizes, int n_in,
                              void* d_out, int out_size, void* d_ws, size_t ws_size,
                              hipStream_t stream) {
  const float* y    = (const float*)d_in[0];  // [8,3,512,512]
  const float* k    = (const float*)d_in[1];  // [8,1,31,31]
  const float* lam  = (const float*)d_in[2];  // scalar
  const float* filt = (const float*)d_in[3];  // [8,3,3,3]
  float* outp = (float*)d_out;                // [8,3,542,542]

  // workspace carve-up (~166 MB of f32)
  float* ws = (float*)d_ws;
  size_t off = 0;
  float* Wr  = ws + off; off += NN;                 // shared DFT matrix (real)
  float* Wi  = ws + off; off += NN;                 // shared DFT matrix (imag)
  float* X   = ws + off; off += (size_t)NIMG * NN;  // padded/tapered image
  float* Tr  = ws + off; off += (size_t)NIMG * NN;  // pass-1 intermediate
  float* Ti  = ws + off; off += (size_t)NIMG * NN;
  float* Fr  = ws + off; off += (size_t)NIMG * NN;  // freq domain / blur scratch
  float* Fi  = ws + off; off += (size_t)NIMG * NN;
  float* Dkr = ws + off; off += (size_t)BB * NN;    // PSF OTF
  float* Dki = ws + off; off += (size_t)BB * NN;
  float* Dg  = ws + off; off += (size_t)CC * NN;    // sum_f |Dg|^2 (no exp(lam))
  float* vtab = ws + off; off += (size_t)BB * 2 * NPAD;
  (void)in_sizes; (void)n_in; (void)out_size; (void)ws_size;

  const int TPB = 256;
  const int nBig = (NIMG * NN + TPB - 1) / TPB;
  const float invN = 1.f / (float)NPAD;

  // 1) DFT matrix, padded image, taper vectors, PSF image (staged in Fr)
  build_dft_kernel<<<(NN + TPB - 1) / TPB, TPB, 0, stream>>>(Wr, Wi);
  pad_edge_kernel<<<nBig, TPB, 0, stream>>>(y, X);
  taper_vecs_kernel<<<dim3(BB, 2), 32, 0, stream>>>(k, vtab);
  build_kimg_kernel<<<(BB * NN + TPB - 1) / TPB, TPB, 0, stream>>>(k, Fr);

  // 2) Dk = fft2(Kimg) = W * Kimg * W   (real input)
  cgemm(Wr, Wi, 0, Fr, nullptr, NN, Tr, Ti, NN, BB, 1.f, 0, 0, stream);
  cgemm(Tr, Ti, NN, Wr, Wi, 0, Dkr, Dki, NN, BB, 1.f, 0, 0, stream);

  // 3) regularizer spectrum
  dg_sum_kernel<<<(NN + TPB - 1) / TPB, TPB, 0, stream>>>(filt, Dg);

  // 4) edgetaper iterations: x = a*x + (1-a)*Re ifft2(fft2(x) * Dk)
  for (int it = 0; it < 3; ++it) {
    cgemm(Wr, Wi, 0, X, nullptr, NN, Tr, Ti, NN, NIMG, 1.f, 0, 0, stream);   // fwd p1
    cgemm(Tr, Ti, NN, Wr, Wi, 0, Fr, Fi, NN, NIMG, 1.f, 0, 0, stream);       // fwd p2
    mul_otf_kernel<<<nBig, TPB, 0, stream>>>(Fr, Fi, Dkr, Dki);
    cgemm(Wr, Wi, 0, Fr, Fi, NN, Tr, Ti, NN, NIMG, invN, 1, 0, stream);      // inv p1
    cgemm(Tr, Ti, NN, Wr, Wi, 0, Fr, nullptr, NN, NIMG, invN, 0, 1, stream); // inv p2 (real)
    blend_kernel<<<nBig, TPB, 0, stream>>>(X, Fr, vtab);
  }

  // 5) Fy = fft2(x); Wiener filter; out = Re ifft2(.)
  cgemm(Wr, Wi, 0, X, nullptr, NN, Tr, Ti, NN, NIMG, 1.f, 0, 0, stream);
  cgemm(Tr, Ti, NN, Wr, Wi, 0, Fr, Fi, NN, NIMG, 1.f, 0, 0, stream);
  wiener_kernel<<<nBig, TPB, 0, stream>>>(Fr, Fi, Dkr, Dki, Dg, lam);
  cgemm(Wr, Wi, 0, Fr, Fi, NN, Tr, Ti, NN, NIMG, invN, 1, 0, stream);
  cgemm(Tr, Ti, NN, Wr, Wi, 0, outp, nullptr, NN, NIMG, invN, 0, 1, stream);
}